// GCNClassifier_71468255805480
// MI455X (gfx1250) — compile-verified
//
#include <hip/hip_runtime.h>

typedef __attribute__((ext_vector_type(2))) float v2f;
typedef __attribute__((ext_vector_type(8))) float v8f;

#define NNODES 50000
#define NEDGES 800000
#define DH     64

// ---------------------------------------------------------------- utilities
__global__ void fill_f32(float* __restrict__ p, float v, int n) {
    int i = blockIdx.x * blockDim.x + threadIdx.x;
    if (i < n) p[i] = v;
}

// deg[dst[e]] += 1  (deg buffer pre-filled with 1.0 for the self-loop)
__global__ void deg_kernel(const int* __restrict__ dst, float* __restrict__ deg, int e) {
    int i = blockIdx.x * blockDim.x + threadIdx.x;
    if (i < e) atomicAdd(&deg[dst[i]], 1.0f);
}

// dis[i] = rsqrt(deg[i])   (deg >= 1 always because of self-loops)
__global__ void rsqrt_kernel(float* __restrict__ d, int n) {
    int i = blockIdx.x * blockDim.x + threadIdx.x;
    if (i < n) d[i] = __frsqrt_rn(d[i]);
}

// ----------------------------------------------------- dense: H[N,64] @ W[64,64]
// One wave32 computes a 16x64 output slab with V_WMMA_F32_16X16X4_F32.
// W staged in LDS pair-interleaved: Wl[(k>>1)*128 + n*2 + (k&1)], so each
// lane's B fragment {W[kb][n], W[kb+1][n]} is a single contiguous ds_load_b64
// landing directly in an even-aligned VGPR pair (no mov shuffles).
__global__ void gemm64_wmma(const float* __restrict__ Hin,
                            const float* __restrict__ W,
                            float* __restrict__ Hout,
                            int ntiles) {
    __shared__ float Wl[32 * 128];            // 16 KB
    for (int i = threadIdx.x; i < DH * DH; i += blockDim.x) {
        int k = i >> 6, n = i & 63;
        Wl[(k >> 1) * 128 + n * 2 + (k & 1)] = W[i];
    }
    __syncthreads();

    const int lane = threadIdx.x & 31;
    const int wave = threadIdx.x >> 5;
    const int tile = blockIdx.x * (blockDim.x >> 5) + wave;
    if (tile >= ntiles) return;               // whole-wave exit: EXEC all-1s for WMMA

    const int m0    = tile * 16;
    const int row   = m0 + (lane & 15);       // A-matrix row for this lane
    const int khalf = (lane >> 4) * 2;        // K sub-offset: 0 (lanes 0-15) / 2 (16-31)
    const int ncol  = lane & 15;              // B/C/D column within 16-wide tile

    v8f acc0 = {}, acc1 = {}, acc2 = {}, acc3 = {};

    #pragma unroll
    for (int k0 = 0; k0 < DH; k0 += 4) {
        const int kb = k0 + khalf;            // even -> 8B aligned
        v2f a = *(const v2f*)(Hin + row * DH + kb);

        const float* bp = &Wl[(kb >> 1) * 128 + ncol * 2];
        v2f b0 = *(const v2f*)(bp +  0);      // cols ncol+0
        v2f b1 = *(const v2f*)(bp + 32);      // cols ncol+16
        v2f b2 = *(const v2f*)(bp + 64);      // cols ncol+32
        v2f b3 = *(const v2f*)(bp + 96);      // cols ncol+48

        acc0 = __builtin_amdgcn_wmma_f32_16x16x4_f32(false, a, false, b0, (short)0, acc0, false, false);
        acc1 = __builtin_amdgcn_wmma_f32_16x16x4_f32(false, a, false, b1, (short)0, acc1, false, false);
        acc2 = __builtin_amdgcn_wmma_f32_16x16x4_f32(false, a, false, b2, (short)0, acc2, false, false);
        acc3 = __builtin_amdgcn_wmma_f32_16x16x4_f32(false, a, false, b3, (short)0, acc3, false, false);
    }

    // C/D layout: VGPR r holds M=r (lanes 0-15) and M=r+8 (lanes 16-31), N=lane%16
    const int rowoff = (lane >> 4) * 8;
    #pragma unroll
    for (int r = 0; r < 8; ++r) {
        float* o = Hout + (m0 + rowoff + r) * DH + ncol;
        o[ 0] = acc0[r];
        o[16] = acc1[r];
        o[32] = acc2[r];
        o[48] = acc3[r];
    }
}

// ------------------------------------------- sparse: A[dst] += B[src] * norm
// 16 threads per edge, 4 consecutive floats each (coalesced atomics).
__global__ void scatter_kernel(const int* __restrict__ src,
                               const int* __restrict__ dst,
                               const float* __restrict__ dis,
                               const float* __restrict__ B,
                               float* __restrict__ A, int e) {
    int tid  = blockIdx.x * blockDim.x + threadIdx.x;
    int eidx = tid >> 4;
    if (eidx >= e) return;
    int f4 = (tid & 15) * 4;
    int s = src[eidx], d = dst[eidx];
    float w = dis[s] * dis[d];
    float4 v = *(const float4*)(B + s * DH + f4);
    float* o = A + d * DH + f4;
    atomicAdd(o + 0, v.x * w);
    atomicAdd(o + 1, v.y * w);
    atomicAdd(o + 2, v.z * w);
    atomicAdd(o + 3, v.w * w);
}

// A = relu(A + B*dis^2 (self-loop) + bias)
__global__ void bias_relu_kernel(float* __restrict__ A,
                                 const float* __restrict__ B,
                                 const float* __restrict__ dis,
                                 const float* __restrict__ bias, int n64) {
    int tid = blockIdx.x * blockDim.x + threadIdx.x;
    if (tid >= n64) return;
    int node = tid >> 6, f = tid & 63;
    float dd = dis[node];
    float v = A[tid] + B[tid] * dd * dd + bias[f];
    A[tid] = v > 0.0f ? v : 0.0f;
}

// out[N,8] = A[N,64] @ linW[64,8] + linb
__global__ void final_linear(const float* __restrict__ A,
                             const float* __restrict__ W,
                             const float* __restrict__ b,
                             float* __restrict__ out, int n8) {
    int tid = blockIdx.x * blockDim.x + threadIdx.x;
    if (tid >= n8) return;
    int nrow = tid >> 3, k = tid & 7;
    const float* a = A + nrow * DH;
    float s = b[k];
    #pragma unroll
    for (int j = 0; j < DH; ++j) s += a[j] * W[j * 8 + k];
    out[tid] = s;
}

extern "C" void kernel_launch(void* const* d_in, const int* in_sizes, int n_in,
                              void* d_out, int out_size, void* d_ws, size_t ws_size,
                              hipStream_t stream) {
    const float* x    = (const float*)d_in[0];
    const int*   edge = (const int*)d_in[1];
    const int*   src  = edge;             // edge_index[0]
    const int*   dst  = edge + NEDGES;    // edge_index[1]
    const float* Ws[3] = { (const float*)d_in[2], (const float*)d_in[4], (const float*)d_in[6] };
    const float* bs[3] = { (const float*)d_in[3], (const float*)d_in[5], (const float*)d_in[7] };
    const float* linW = (const float*)d_in[8];
    const float* linb = (const float*)d_in[9];

    // workspace: A[N*64] | B[N*64] | dis[N]
    float* A   = (float*)d_ws;
    float* B   = A + NNODES * DH;
    float* dis = B + NNODES * DH;

    const int n64 = NNODES * DH;

    // --- symmetric normalization: deg (with self-loop) -> rsqrt
    fill_f32<<<(NNODES + 255) / 256, 256, 0, stream>>>(dis, 1.0f, NNODES);
    deg_kernel<<<(NEDGES + 255) / 256, 256, 0, stream>>>(dst, dis, NEDGES);
    rsqrt_kernel<<<(NNODES + 255) / 256, 256, 0, stream>>>(dis, NNODES);

    const int ntiles = NNODES / 16;                    // 3125, exact
    const int gblk   = (ntiles + 3) / 4;               // 4 waves / block (128 thr)

    const float* hin = x;
    for (int l = 0; l < 3; ++l) {
        gemm64_wmma<<<gblk, 128, 0, stream>>>(hin, Ws[l], B, ntiles);
        fill_f32<<<(n64 + 255) / 256, 256, 0, stream>>>(A, 0.0f, n64);
        scatter_kernel<<<(NEDGES * 16 + 255) / 256, 256, 0, stream>>>(src, dst, dis, B, A, NEDGES);
        bias_relu_kernel<<<(n64 + 255) / 256, 256, 0, stream>>>(A, B, dis, bs[l], n64);
        hin = A;
    }

    final_linear<<<(NNODES * 8 + 255) / 256, 256, 0, stream>>>(A, linW, linb, (float*)d_out, NNODES * 8);
}